// QuantumLayer_23278722744719
// MI455X (gfx1250) — compile-verified
//
#include <hip/hip_runtime.h>

// ============================================================================
// 21-qubit real state-vector sim on MI455X (gfx1250).
// Strategy: fuse RY gates 4-qubits-at-a-time into 16x16 matrices applied with
// V_WMMA_F32_16X16X4_F32 (f32 in/out, 4 K-chunks per 16x16 unitary).
// CNOT ring (a GF(2)-linear index permutation) is folded into the gather
// addresses of the first WMMA pass of each layer (ping-pong buffers).
// State (32 MB) fits in the 192 MB L2, so all 24 passes run at L2 bandwidth.
// ============================================================================

typedef __attribute__((ext_vector_type(2))) float v2f;
typedef __attribute__((ext_vector_type(8))) float v8f;

#define NQ      21
#define NSTATE  (1u << NQ)
#define BATCH   4
#define NGROUP  6          // 5 groups of 4 qubits + {17,18,19,20} w/ RY only on 20
#define NLAYERS 3
#define TILES   8          // position-tiles per wave (A fragments reused)

// Qubit q occupies bit (20 - q) of the state index (reference reshape order).

// Inverse index map of the full 21-CNOT ring (applied C20 first ... C0 last).
// Each step is linear over GF(2), so the whole map is linear: perm(a^b)=perm(a)^perm(b).
__device__ __forceinline__ unsigned perm21(unsigned y) {
  y ^= (y & 1u) << 20;                 // i=20: CNOT(q20 -> q0): cbit 0, tbit 20
#pragma unroll
  for (int i = 19; i >= 0; --i)        // i: CNOT(qi -> qi+1): cbit 20-i, tbit 19-i
    y ^= ((y >> (20 - i)) & 1u) << (19 - i);
  return y;
}

// ---------------------------------------------------------------------------
// Build the 42 fused 16x16 group matrices.
//   m in [0,24):  initial layer, per-batch angles x[b][q]   -> mats[(b*6+gi)]
//   m in [24,42): weight layer l, shared angles w[l][q]     -> mats[24+l*6+gi]
// Group gi<5 covers qubits 4gi..4gi+3 (all active). gi==5 covers qubits 17..20
// with identity on 17,18,19 and RY on qubit 20. U = M_q0 (x) M_q0+1 (x) ...
// (q0 = high bit of the 4-bit group index).  M = [[c,-s],[s,c]], c=cos(t/2).
// ---------------------------------------------------------------------------
__global__ void build_mats(const float* __restrict__ x,
                           const float* __restrict__ w,
                           float* __restrict__ mats) {
  int m = blockIdx.x;                 // 0..41
  int t = threadIdx.x;                // 0..255
  int gi;
  const float* ang;
  if (m < 24) { gi = m % NGROUP; ang = x + (m / NGROUP) * NQ; }
  else        { gi = (m - 24) % NGROUP; ang = w + ((m - 24) / NGROUP) * NQ; }
  int q0 = (gi < 5) ? 4 * gi : 17;
  int g = t >> 4, gp = t & 15;
  float val = 1.0f;
#pragma unroll
  for (int k = 0; k < 4; ++k) {
    int bg = (g  >> (3 - k)) & 1;     // qubit q0+k <-> group bit (3-k)
    int bp = (gp >> (3 - k)) & 1;
    bool active = (gi < 5) || (k == 3);
    float f;
    if (active) {
      float th = 0.5f * ang[q0 + k];
      float c = cosf(th), s = sinf(th);
      f = bg ? (bp ? c : s) : (bp ? -s : c);   // [[c,-s],[s,c]]
    } else {
      f = (bg == bp) ? 1.0f : 0.0f;            // identity qubit
    }
    val *= f;
  }
  mats[m * 256 + g * 16 + gp] = val;
}

__global__ void init_state(float* __restrict__ s) {
  unsigned gid = blockIdx.x * blockDim.x + threadIdx.x;   // over BATCH*NSTATE
  s[gid] = ((gid & (NSTATE - 1u)) == 0u) ? 1.0f : 0.0f;
}

// ---------------------------------------------------------------------------
// Apply a 16x16 fused matrix U to qubit group occupying bits [sh, sh+4).
// State index = (hi << (sh+4)) | (g << sh) | lo.  Each wave processes TILES
// tiles of 16 positions; per tile: 8 gather loads/lane -> 4x WMMA f32 16x16x4
// (K-chunks of the 16x16 unitary) -> 8 stores/lane.
// doPerm: fold the CNOT-ring inverse permutation into the gather addresses
// (must run out-of-place; dst != src). Otherwise in-place is safe: each wave
// owns disjoint (hi,lo) fibers.
// ---------------------------------------------------------------------------
__global__ void apply_group16(float* __restrict__ dst, const float* __restrict__ src,
                              const float* __restrict__ mats, int matBatchStride,
                              int sh, int doPerm) {
  const int b = blockIdx.y;
  const float* U = mats + (size_t)b * matBatchStride;
  const unsigned lane = threadIdx.x & 31u;
  const unsigned col  = lane & 15u;          // N (position) / A-matrix M row
  const unsigned half = lane >> 4;           // K half-wave split
  const unsigned wave = blockIdx.x * (blockDim.x >> 5) + (threadIdx.x >> 5);

  const float* sb = src + ((size_t)b << NQ);
  float*       db = dst + ((size_t)b << NQ);

  // A fragments (16x4 f32): VGPR0 = K k0 / k0+2, VGPR1 = K k0+1 / k0+3.
  v2f a[4];
#pragma unroll
  for (int c = 0; c < 4; ++c) {
    int k0 = 4 * c + 2 * (int)half;
    a[c].x = U[col * 16 + k0];
    a[c].y = U[col * 16 + k0 + 1];
  }

  // Linear-permutation basis masks for the 4 group bits (wave-uniform).
  unsigned m0 = 0, m1 = 0, m2 = 0, m3 = 0;
  if (doPerm) {
    m0 = perm21(1u << sh); m1 = perm21(2u << sh);
    m2 = perm21(4u << sh); m3 = perm21(8u << sh);
  }
  const unsigned loMask = (1u << sh) - 1u;

  for (int t = 0; t < TILES; ++t) {
    unsigned p   = wave * (16u * TILES) + 16u * (unsigned)t + col; // position
    unsigned lo  = p & loMask;
    unsigned hi  = p >> sh;
    unsigned base = (hi << (sh + 4)) | lo;   // index with g = 0

    // B fragments (4x16 f32): lane half selects K+2; VGPR v -> K = 4c+v+2*half.
    v2f bf[4];
    if (doPerm) {
      unsigned pb = perm21(base);            // perm of non-group bits
#pragma unroll
      for (int c = 0; c < 4; ++c) {
        unsigned g0 = 4u * c + 2u * half;    // even
        unsigned gm0 = ((g0 & 2u) ? m1 : 0u) ^ ((g0 & 4u) ? m2 : 0u) ^
                       ((g0 & 8u) ? m3 : 0u);
        bf[c].x = sb[pb ^ gm0];
        bf[c].y = sb[pb ^ gm0 ^ m0];         // g0+1 = g0 ^ 1
      }
    } else {
#pragma unroll
      for (int c = 0; c < 4; ++c) {
        unsigned g0 = 4u * c + 2u * half;
        bf[c].x = sb[base | (g0 << sh)];
        bf[c].y = sb[base | ((g0 + 1u) << sh)];
      }
    }

    // out[g] = sum_{g'} U[g][g'] * in[g'] for 16 positions, via 4 K-chunks.
    v8f acc = {};
#pragma unroll
    for (int c = 0; c < 4; ++c)
      acc = __builtin_amdgcn_wmma_f32_16x16x4_f32(
          /*neg_a=*/false, a[c], /*neg_b=*/false, bf[c],
          /*c_mod=*/(short)0, acc, /*reuse_a=*/false, /*reuse_b=*/false);

    // D layout: VGPR v -> M = v + 8*half, N = col.
#pragma unroll
    for (int v = 0; v < 8; ++v) {
      unsigned g = (unsigned)v + 8u * half;
      db[base | (g << sh)] = acc[v];
    }
  }
}

// ---------------------------------------------------------------------------
// Deterministic two-stage <Z_q> reduction: z[b][q] = sum probs * (1 - 2*bit).
// ---------------------------------------------------------------------------
__global__ void reduce_z(const float* __restrict__ s, float* __restrict__ part) {
  const int b = blockIdx.y, chunk = blockIdx.x, tid = threadIdx.x;
  const float* sb = s + ((size_t)b << NQ);
  const unsigned span = NSTATE / 256u;       // 8192 elements per chunk
  const unsigned baseE = (unsigned)chunk * span;
  float acc[NQ];
#pragma unroll
  for (int q = 0; q < NQ; ++q) acc[q] = 0.0f;
  for (unsigned e = baseE + tid; e < baseE + span; e += 256u) {
    float v = sb[e];
    float pr = v * v;
#pragma unroll
    for (int q = 0; q < NQ; ++q)
      acc[q] += ((e >> (20 - q)) & 1u) ? -pr : pr;
  }
  __shared__ float red[256];
  for (int q = 0; q < NQ; ++q) {
    red[tid] = acc[q];
    __syncthreads();
    for (int off = 128; off > 0; off >>= 1) {
      if (tid < off) red[tid] += red[tid + off];
      __syncthreads();
    }
    if (tid == 0) part[((size_t)(b * 256 + chunk)) * NQ + q] = red[0];
    __syncthreads();
  }
}

__global__ void finalize_z(const float* __restrict__ part, float* __restrict__ out) {
  int t = blockIdx.x * blockDim.x + threadIdx.x;
  if (t >= BATCH * NQ) return;
  int b = t / NQ, q = t % NQ;
  float s = 0.0f;
  for (int c = 0; c < 256; ++c) s += part[((size_t)(b * 256 + c)) * NQ + q];
  out[t] = s;
}

// ---------------------------------------------------------------------------
extern "C" void kernel_launch(void* const* d_in, const int* in_sizes, int n_in,
                              void* d_out, int out_size, void* d_ws, size_t ws_size,
                              hipStream_t stream) {
  const float* x = (const float*)d_in[0];   // (4, 21)
  const float* w = (const float*)d_in[1];   // (3, 21)
  float* out = (float*)d_out;               // (4, 21)

  // Workspace: two state buffers (ping-pong) + fused matrices + partials.
  float* S0   = (float*)d_ws;                         // 4 * 2^21 floats
  float* S1   = S0 + (size_t)BATCH * NSTATE;          // 4 * 2^21 floats
  float* MAT  = S1 + (size_t)BATCH * NSTATE;          // 42 * 256 floats
  float* PART = MAT + 42 * 256;                       // 4 * 256 * 21 floats

  build_mats<<<42, 256, 0, stream>>>(x, w, MAT);
  init_state<<<(BATCH * NSTATE) / 256, 256, 0, stream>>>(S0);

  // Group low-bit shifts: q0 = {0,4,8,12,16,17} -> sh = 17 - q0 (bit = 20-q).
  const int SHS[NGROUP] = {17, 13, 9, 5, 1, 0};
  dim3 grid(128, BATCH);                    // 1024 waves/batch * 128 positions

  float* cur = S0;
  float* alt = S1;

  // Initial RY layer (per-batch matrices, stride 6*256 floats), in-place.
  for (int gi = 0; gi < NGROUP; ++gi)
    apply_group16<<<grid, 256, 0, stream>>>(cur, cur, MAT + gi * 256,
                                            NGROUP * 256, SHS[gi], 0);

  // 3 x (CNOT ring folded into pass 0 gather, then 5 in-place RY passes).
  for (int l = 0; l < NLAYERS; ++l) {
    float* lm = MAT + (24 + l * NGROUP) * 256;
    apply_group16<<<grid, 256, 0, stream>>>(alt, cur, lm, 0, SHS[0], 1);
    float* tmp = cur; cur = alt; alt = tmp;
    for (int gi = 1; gi < NGROUP; ++gi)
      apply_group16<<<grid, 256, 0, stream>>>(cur, cur, lm + gi * 256,
                                              0, SHS[gi], 0);
  }

  reduce_z<<<dim3(256, BATCH), 256, 0, stream>>>(cur, PART);
  finalize_z<<<1, 128, 0, stream>>>(PART, out);
}